// ScanMerge_36206574305366
// MI455X (gfx1250) — compile-verified
//
#include <hip/hip_runtime.h>

// Problem constants from the reference
#define BB 32
#define NN 1024
#define CC 768
#define HH 32

typedef __attribute__((ext_vector_type(2))) float v2f;
typedef __attribute__((ext_vector_type(8))) float v8f;

// ---------------------------------------------------------------------------
// Kernel 0: zero the workspace accumulators (pooled sums + alpha)
// ---------------------------------------------------------------------------
__global__ void init_ws_kernel(float* ws, int n) {
    for (int i = threadIdx.x; i < n; i += blockDim.x) ws[i] = 0.0f;
}

// ---------------------------------------------------------------------------
// Kernel 1: pooled[b][c] = sum_n x[b][n][c]   (mean scale folded in later)
// Grid: (B=32, 32 n-chunks of 32 rows); block: 192 threads, one float4 per
// thread covers a full 768-float row -> global_load_b128 streaming reads.
// Streams all of x through L2 (100 MB < 192 MB) ahead of the merge pass.
// ---------------------------------------------------------------------------
__global__ void __launch_bounds__(192)
pool_kernel(const float* __restrict__ x, float* __restrict__ pooled) {
    const int b  = blockIdx.x;
    const int n0 = blockIdx.y * 32;
    const int t4 = threadIdx.x;                    // 0..191 float4 index
    const int R  = CC / 4;
    const float4* p = (const float4*)x + ((size_t)b * NN + n0) * R + t4;
    float4 acc = {0.f, 0.f, 0.f, 0.f};
#pragma unroll 4
    for (int n = 0; n < 32; ++n) {
        float4 v = p[(size_t)n * R];
        acc.x += v.x; acc.y += v.y; acc.z += v.z; acc.w += v.w;
    }
    float* dst = &pooled[b * CC + t4 * 4];
    atomicAdd(dst + 0, acc.x);                     // global_atomic_add_f32
    atomicAdd(dst + 1, acc.y);
    atomicAdd(dst + 2, acc.z);
    atomicAdd(dst + 3, acc.w);
}

// ---------------------------------------------------------------------------
// Kernel 2: alpha[b][k] = (pooled[b,:]/1024) . W[k,:] + bias[k]
// 32x4x768 fp32 GEMM via V_WMMA_F32_16X16X4_F32.
// One block of 64 threads = 2 waves; wave w owns batch rows 16w..16w+15.
// A (16x4 f32, 2 VGPRs/lane):  lane<16 -> M=lane, K={k0,k0+1}
//                              lane>=16 -> M=lane-16, K={k0+2,k0+3}
// B (4x16 f32, 2 VGPRs/lane):  N=lane%16, same K split; B[k][n]=W[n][k], n>=4 -> 0
// D (16x16 f32, 8 VGPRs): VGPR r -> M = r + 8*(lane/16), N = lane%16
// No divergent branches before/inside the WMMA loop (EXEC stays all-ones).
// ---------------------------------------------------------------------------
__global__ void alpha_kernel(const float* __restrict__ pooled,
                             const float* __restrict__ W,
                             const float* __restrict__ bias,
                             float* __restrict__ alpha) {
    const int lane = threadIdx.x & 31;
    const int wv   = threadIdx.x >> 5;     // 0 or 1: which 16-row batch tile
    const int half = lane >> 4;            // 0: lanes 0-15, 1: lanes 16-31
    const int lid  = lane & 15;

    const int m      = 16 * wv + lid;      // A-matrix row this lane feeds
    const int nclamp = (lid < 4) ? lid : 3;
    const float nmask = (lid < 4) ? 1.0f : 0.0f;   // select, not branch
    const float scale = 1.0f / 1024.0f;

    v8f acc = {};
    for (int k0 = 0; k0 < CC; k0 += 4) {
        const int k = k0 + 2 * half;
        v2f a, bm;
        a.x = pooled[m * CC + k + 0] * scale;
        a.y = pooled[m * CC + k + 1] * scale;
        bm.x = W[nclamp * CC + k + 0] * nmask;
        bm.y = W[nclamp * CC + k + 1] * nmask;
        acc = __builtin_amdgcn_wmma_f32_16x16x4_f32(
            /*neg_a=*/false, a, /*neg_b=*/false, bm,
            /*c_mod=*/(short)0, acc, /*reuse_a=*/false, /*reuse_b=*/false);
    }

    if (lid < 4) {
        const float bk = bias[lid];
#pragma unroll
        for (int r = 0; r < 8; ++r) {
            const int mr = 16 * wv + r + 8 * half;   // D-layout row
            alpha[mr * 4 + lid] = acc[r] + bk;
        }
    }
}

// ---------------------------------------------------------------------------
// Kernel 3: grouped merge. The index set {n, t(n), r(n), t(r(n))} is closed
// (t=32x32 transpose, r=reverse), so one block loads the 4 input rows ONCE
// and emits all 4 output rows with permuted coefficients:
//   out[n]  = a0 v0 + a1 v1 + a2 v2 + a3 v3
//   out[t]  = a0 v1 + a1 v0 + a2 v3 + a3 v2
//   out[r]  = a0 v2 + a1 v3 + a2 v0 + a3 v1
//   out[tr] = a0 v3 + a1 v2 + a2 v1 + a3 v0
// Leaders: h<w (496 4-groups) and h==w<16 (16 diagonal 2-groups; the
// duplicated rows get bit-identical double writes). Other blocks exit.
// 192 threads x float4 = one 768-float row per load/store (B128 ops).
// ---------------------------------------------------------------------------
__device__ __forceinline__ float4 comb4(float a0, const float4& p,
                                        float a1, const float4& q,
                                        float a2, const float4& s,
                                        float a3, const float4& t) {
    float4 r;
    r.x = a0 * p.x + a1 * q.x + a2 * s.x + a3 * t.x;
    r.y = a0 * p.y + a1 * q.y + a2 * s.y + a3 * t.y;
    r.z = a0 * p.z + a1 * q.z + a2 * s.z + a3 * t.z;
    r.w = a0 * p.w + a1 * q.w + a2 * s.w + a3 * t.w;
    return r;
}

__global__ void __launch_bounds__(192)
merge_kernel(const float* __restrict__ x, const float* __restrict__ alpha,
             float* __restrict__ out) {
    const int h = blockIdx.x & 31;
    const int w = blockIdx.x >> 5;
    const int b = blockIdx.y;

    // group-leader test: upper triangle, or first half of the diagonal
    if (h > w) return;
    if (h == w && h >= 16) return;

    const int n  = (h << 5) | w;                    // identity
    const int n1 = (w << 5) | h;                    // transpose
    const int n2 = (NN - 1) - n;                    // reverse
    const int n3 = (NN - 1) - n1;                   // transpose of reverse

    const int t4 = threadIdx.x;                     // 0..191
    const int R  = CC / 4;
    const float4* xb = (const float4*)(x + (size_t)b * NN * CC);

    float4 v0 = xb[(size_t)n  * R + t4];
    float4 v1 = xb[(size_t)n1 * R + t4];
    float4 v2 = xb[(size_t)n2 * R + t4];
    float4 v3 = xb[(size_t)n3 * R + t4];

    const float a0 = alpha[b * 4 + 0];
    const float a1 = alpha[b * 4 + 1];
    const float a2 = alpha[b * 4 + 2];
    const float a3 = alpha[b * 4 + 3];

    float4* ob = (float4*)(out + (size_t)b * NN * CC);
    ob[(size_t)n  * R + t4] = comb4(a0, v0, a1, v1, a2, v2, a3, v3);
    ob[(size_t)n1 * R + t4] = comb4(a0, v1, a1, v0, a2, v3, a3, v2);
    ob[(size_t)n2 * R + t4] = comb4(a0, v2, a1, v3, a2, v0, a3, v1);
    ob[(size_t)n3 * R + t4] = comb4(a0, v3, a1, v2, a2, v1, a3, v0);
}

// ---------------------------------------------------------------------------
extern "C" void kernel_launch(void* const* d_in, const int* in_sizes, int n_in,
                              void* d_out, int out_size, void* d_ws, size_t ws_size,
                              hipStream_t stream) {
    const float* x    = (const float*)d_in[0];   // (32, 1024, 768) f32
    const float* W    = (const float*)d_in[1];   // (4, 768) f32
    const float* bias = (const float*)d_in[2];   // (4,) f32
    float* out = (float*)d_out;                  // (32, 1024, 768) f32

    float* pooled = (float*)d_ws;                // 32*768 accumulators
    float* alpha  = pooled + BB * CC;            // 32*4

    init_ws_kernel<<<1, 256, 0, stream>>>(pooled, BB * CC + BB * 4);
    pool_kernel<<<dim3(BB, 32), 192, 0, stream>>>(x, pooled);
    alpha_kernel<<<1, 64, 0, stream>>>(pooled, W, bias, alpha);
    merge_kernel<<<dim3(NN, BB), 192, 0, stream>>>(x, alpha, out);
}